// MultiHeadAttention_61022895341644
// MI455X (gfx1250) — compile-verified
//
#include <hip/hip_runtime.h>

// ---------------------------------------------------------------------------
// MultiHeadAttention (no-softmax variant, per reference) for MI455X / gfx1250
//   B=2, S=2048, D_MODEL=1024, H=16, DK=64, all fp32.
//   All four GEMM stages + fused attention middle use V_WMMA_F32_16X16X4_F32.
//   v2: 32-row wave tiles everywhere -> fewer loads per WMMA (10.7 FLOP/B
//   register-level intensity in the GEMMs, shared K/V operand loads in attn).
// ---------------------------------------------------------------------------

typedef float v2f __attribute__((ext_vector_type(2)));
typedef float v8f __attribute__((ext_vector_type(8)));

#define DMODEL 1024
#define NHEAD  16
#define DK     64
#define NEG_INF (-1.0e9f)

__device__ inline v8f wmma_f32(v2f a, v2f b, v8f c) {
  // (neg_a, A, neg_b, B, c_mod, C, reuse_a, reuse_b)
  return __builtin_amdgcn_wmma_f32_16x16x4_f32(false, a, false, b,
                                               (short)0, c, false, false);
}

// ---------------------------------------------------------------------------
// Y[M,N] = X[M,K] @ W[N,K]^T + bias[N]        (NT GEMM, fp32 WMMA)
// One wave computes a 32x64 tile: 2 M-subtiles x 4 N-subtiles, 8 accums.
// Per K-step (K+=4): 2 A-loads + 4 B-loads feed 8 WMMAs (all float2/lane).
// Block = 8 waves stacked in M -> 256x64 macro tile.
// qkv_layout != 0: scatter output to [B, H, S, DK] panels.
// ---------------------------------------------------------------------------
__global__ void __launch_bounds__(256)
gemm_nt_bias_kernel(const float* __restrict__ X, const float* __restrict__ W,
                    const float* __restrict__ bias, float* __restrict__ Y,
                    int M, int N, int K, int S, int qkv_layout) {
  const int lane = threadIdx.x & 31;
  const int wave = threadIdx.x >> 5;
  const int half = lane >> 4;          // 0 or 1
  const int l    = lane & 15;          // 0..15
  const int mBase = (blockIdx.y * 8 + wave) * 32;
  const int nBase = blockIdx.x * 64;

  const float* xrow0 = X + (size_t)(mBase +  0 + l) * K;
  const float* xrow1 = X + (size_t)(mBase + 16 + l) * K;
  const float* w0    = W + (size_t)(nBase +  0 + l) * K;
  const float* w1    = W + (size_t)(nBase + 16 + l) * K;
  const float* w2    = W + (size_t)(nBase + 32 + l) * K;
  const float* w3    = W + (size_t)(nBase + 48 + l) * K;

  v8f acc[2][4] = {{v8f{}, v8f{}, v8f{}, v8f{}},
                   {v8f{}, v8f{}, v8f{}, v8f{}}};

  #pragma unroll 4
  for (int k0 = 0; k0 < K; k0 += 4) {
    const int off = k0 + 2 * half;
    v2f a0 = *(const v2f*)(xrow0 + off);
    v2f a1 = *(const v2f*)(xrow1 + off);
    v2f b0 = *(const v2f*)(w0 + off);
    v2f b1 = *(const v2f*)(w1 + off);
    v2f b2 = *(const v2f*)(w2 + off);
    v2f b3 = *(const v2f*)(w3 + off);
    acc[0][0] = wmma_f32(a0, b0, acc[0][0]);
    acc[0][1] = wmma_f32(a0, b1, acc[0][1]);
    acc[0][2] = wmma_f32(a0, b2, acc[0][2]);
    acc[0][3] = wmma_f32(a0, b3, acc[0][3]);
    acc[1][0] = wmma_f32(a1, b0, acc[1][0]);
    acc[1][1] = wmma_f32(a1, b1, acc[1][1]);
    acc[1][2] = wmma_f32(a1, b2, acc[1][2]);
    acc[1][3] = wmma_f32(a1, b3, acc[1][3]);
  }

  #pragma unroll
  for (int mi = 0; mi < 2; ++mi) {
    #pragma unroll
    for (int j = 0; j < 4; ++j) {
      const int n  = nBase + j * 16 + l;
      const float bv = bias[n];
      #pragma unroll
      for (int r = 0; r < 8; ++r) {
        const int m = mBase + mi * 16 + r + 8 * half;
        const float v = acc[mi][j][r] + bv;
        size_t dst;
        if (qkv_layout) {
          const int bidx = m / S, s = m % S;
          const int h = n / DK,   d = n % DK;
          dst = (((size_t)(bidx * NHEAD + h) * S + s) * DK + d);
        } else {
          dst = (size_t)m * N + n;
        }
        Y[dst] = v;
      }
    }
  }
}

// ---------------------------------------------------------------------------
// Fused attention middle (NO softmax, per reference):
//   S_tile = (Q_tile · K_tile^T) / 8 ; masked with -1e9 where mask==0
//   O_tile += S_tile · V_tile
// One wave owns 32 q-rows (two 16-row S-tiles, 8 O accumulators); the K-panel
// loads are shared by both score tiles and the V loads by both output rows.
// Block of 8 waves covers 256 q rows. Q/K/V panels are [S, DK] per (b,h).
// Output written to [B, S, DMODEL] so the final projection is a plain GEMM.
// ---------------------------------------------------------------------------
__global__ void __launch_bounds__(256)
attn_nosoftmax_kernel(const float* __restrict__ Q, const float* __restrict__ Kp,
                      const float* __restrict__ V, const int* __restrict__ mask,
                      float* __restrict__ O, int S) {
  const int lane = threadIdx.x & 31;
  const int wave = threadIdx.x >> 5;
  const int half = lane >> 4;
  const int l    = lane & 15;
  const int bh   = blockIdx.y;             // 0 .. B*H-1
  const int bidx = bh / NHEAD;
  const int h    = bh % NHEAD;
  const int qBase = blockIdx.x * 256 + wave * 32;

  const float* Qh = Q  + (size_t)bh * S * DK;
  const float* Kh = Kp + (size_t)bh * S * DK;
  const float* Vh = V  + (size_t)bh * S * DK;

  __shared__ float lds_s[8][2 * 16 * 16]; // 2 KB per wave: two 16x16 S-tiles
  float* sbuf = lds_s[wave];

  // Preload both 16x64 Q subtiles in A-operand layout (float2/lane per step).
  v2f qa[2][16];
  const float* qrow0 = Qh + (size_t)(qBase +  0 + l) * DK;
  const float* qrow1 = Qh + (size_t)(qBase + 16 + l) * DK;
  #pragma unroll
  for (int kk = 0; kk < 16; ++kk) {
    qa[0][kk] = *(const v2f*)(qrow0 + kk * 4 + 2 * half);
    qa[1][kk] = *(const v2f*)(qrow1 + kk * 4 + 2 * half);
  }

  v8f o[2][4] = {{v8f{}, v8f{}, v8f{}, v8f{}},
                 {v8f{}, v8f{}, v8f{}, v8f{}}};

  for (int kt = 0; kt < S; kt += 16) {
    // ---- two S-tiles = Q · K^T over d = 0..63; K-loads shared ------------
    v8f s0 = {}, s1 = {};
    const float* krow = Kh + (size_t)(kt + l) * DK;
    #pragma unroll
    for (int kk = 0; kk < 16; ++kk) {
      v2f kb = *(const v2f*)(krow + kk * 4 + 2 * half);
      s0 = wmma_f32(qa[0][kk], kb, s0);
      s1 = wmma_f32(qa[1][kk], kb, s1);
    }

    // ---- scale + mask (mask depends only on key column = l) --------------
    const int mval = mask[bidx * S + kt + l];
    #pragma unroll
    for (int r = 0; r < 8; ++r) {
      const float v0 = (mval != 0) ? s0[r] * 0.125f : NEG_INF;
      const float v1 = (mval != 0) ? s1[r] * 0.125f : NEG_INF;
      sbuf[(r + 8 * half) * 16 + l]       = v0;  // tile 0, row-major 16x16
      sbuf[256 + (r + 8 * half) * 16 + l] = v1;  // tile 1
    }
    // Cross-lane C-layout -> A-layout handoff within the (lock-step) wave:
    // only need the DS writes to land before the DS reads below.
    asm volatile("s_wait_dscnt 0" ::: "memory");

    // ---- O += S_tile · V_tile (K=16 -> 4 steps; V loads shared) ----------
    #pragma unroll
    for (int kk2 = 0; kk2 < 4; ++kk2) {
      v2f sa0 = *(const v2f*)(&sbuf[l * 16 + kk2 * 4 + 2 * half]);
      v2f sa1 = *(const v2f*)(&sbuf[256 + l * 16 + kk2 * 4 + 2 * half]);
      const float* vrow = Vh + (size_t)(kt + kk2 * 4 + 2 * half) * DK;
      #pragma unroll
      for (int j = 0; j < 4; ++j) {
        v2f vb;
        vb[0] = vrow[j * 16 + l];          // B[K = k0+2*half    ][N = l]
        vb[1] = vrow[DK + j * 16 + l];     // B[K = k0+2*half + 1][N = l]
        o[0][j] = wmma_f32(sa0, vb, o[0][j]);
        o[1][j] = wmma_f32(sa1, vb, o[1][j]);
      }
    }
  }

  // ---- store O tiles to [B, S, DMODEL] -----------------------------------
  #pragma unroll
  for (int mi = 0; mi < 2; ++mi) {
    #pragma unroll
    for (int j = 0; j < 4; ++j) {
      #pragma unroll
      for (int r = 0; r < 8; ++r) {
        const int q = qBase + mi * 16 + r + 8 * half;
        O[((size_t)bidx * S + q) * DMODEL + h * DK + j * 16 + l] = o[mi][j][r];
      }
    }
  }
}

// ---------------------------------------------------------------------------
// Host-side launcher
// ---------------------------------------------------------------------------
extern "C" void kernel_launch(void* const* d_in, const int* in_sizes, int n_in,
                              void* d_out, int out_size, void* d_ws, size_t ws_size,
                              hipStream_t stream) {
  const int B = 2, S = 2048;
  const int M = B * S;                       // 4096 rows
  (void)in_sizes; (void)n_in; (void)out_size; (void)ws_size;

  const float* q   = (const float*)d_in[0];
  const float* k   = (const float*)d_in[1];
  const float* v   = (const float*)d_in[2];
  const int*   msk = (const int*)  d_in[3];
  const float* W_q = (const float*)d_in[4];
  const float* b_q = (const float*)d_in[5];
  const float* W_k = (const float*)d_in[6];
  const float* b_k = (const float*)d_in[7];
  const float* W_v = (const float*)d_in[8];
  const float* b_v = (const float*)d_in[9];
  const float* W_o = (const float*)d_in[10];
  const float* b_o = (const float*)d_in[11];

  const size_t panel = (size_t)B * NHEAD * S * DK;   // 4M floats = 16 MB
  float* wsQ = (float*)d_ws;
  float* wsK = wsQ + panel;
  float* wsV = wsK + panel;
  float* wsA = wsV + panel;                           // attn_out [B,S,DMODEL]

  dim3 blk(256);
  dim3 gemmGrid(DMODEL / 64, M / 256);                // (16, 16)

  // Q/K/V projections -> [B,H,S,DK] panels
  gemm_nt_bias_kernel<<<gemmGrid, blk, 0, stream>>>(q, W_q, b_q, wsQ,
                                                    M, DMODEL, DMODEL, S, 1);
  gemm_nt_bias_kernel<<<gemmGrid, blk, 0, stream>>>(k, W_k, b_k, wsK,
                                                    M, DMODEL, DMODEL, S, 1);
  gemm_nt_bias_kernel<<<gemmGrid, blk, 0, stream>>>(v, W_v, b_v, wsV,
                                                    M, DMODEL, DMODEL, S, 1);

  // Fused masked-scores x V (no softmax)
  dim3 attnGrid(S / 256, B * NHEAD);                  // (8, 32)
  attn_nosoftmax_kernel<<<attnGrid, blk, 0, stream>>>(wsQ, wsK, wsV, msk,
                                                      wsA, S);

  // Output projection -> d_out [B,S,DMODEL]
  gemm_nt_bias_kernel<<<gemmGrid, blk, 0, stream>>>(wsA, W_o, b_o,
                                                    (float*)d_out,
                                                    M, DMODEL, DMODEL, S, 0);
}